// ShellProvider_66245575573883
// MI455X (gfx1250) — compile-verified
//
#include <hip/hip_runtime.h>
#include <stdint.h>

// Problem constants from the reference: B=16, A=2048, N=512
#define NB_B 16
#define NB_A 2048
#define NB_N 512
#define APB  16          // atoms handled per block
#define TPB  256         // 8 wave32s per block
#define EPS_F 1e-8f

typedef __attribute__((ext_vector_type(4))) float f32x4;
typedef __attribute__((ext_vector_type(4))) int   i32x4;

// Address-space-qualified pointer types for the async-DMA builtin
// (signature: (v4i addrspace(1)* src, v4i addrspace(3)* dst, imm off, imm cpol))
typedef __attribute__((address_space(1))) i32x4 gi32x4;
typedef __attribute__((address_space(3))) i32x4 li32x4;

// 1-ulp native transcendentals (TRANS32 ops: co-execute with VALU on CDNA5)
static __device__ __forceinline__ float fast_sqrt(float x) {
#if __has_builtin(__builtin_amdgcn_sqrtf)
  return __builtin_amdgcn_sqrtf(x);      // v_sqrt_f32
#else
  return sqrtf(x);
#endif
}
static __device__ __forceinline__ float fast_rcp(float x) {
#if __has_builtin(__builtin_amdgcn_rcpf)
  return __builtin_amdgcn_rcpf(x);       // v_rcp_f32
#else
  return 1.0f / x;
#endif
}

// D = distances (B,A,N); V = distance vectors (B,A,N,3);
// pass-throughs: neighbors (B,A,N) i32, mask (B,A,N) i32.
// d_out layout (flat, return order): [D | V | neighbors | mask]
__global__ __launch_bounds__(TPB) void neighbor_dist_kernel(
    const float* __restrict__ atoms,
    const int*   __restrict__ neighbors,
    const int*   __restrict__ nmask,
    float* __restrict__ out)
{
  // One batch's full atom table: 2048 atoms * 3 f32 = 24 KB (LDS budget 320 KB/WGP)
  __shared__ float s_atoms[NB_A * 3];

  const int tid   = threadIdx.x;
  const int bpb   = NB_A / APB;              // blocks per batch = 128
  const int b     = blockIdx.x / bpb;
  const int chunk = blockIdx.x % bpb;

  const float* atoms_b = atoms + (size_t)b * (NB_A * 3);

  // ---- Stage atoms[b] into LDS via CDNA5 async global->LDS DMA ----
#if __has_builtin(__builtin_amdgcn_global_load_async_to_lds_b128)
  #pragma unroll
  for (int i = 0; i < (NB_A * 3) / (TPB * 4); ++i) {   // 6 rounds of 4 KB
    const int f = (i * TPB + tid) * 4;                 // float offset, 16B granules
    gi32x4* gsrc = (gi32x4*)(uintptr_t)(atoms_b + f);
    li32x4* ldst = (li32x4*)(uint32_t)(uintptr_t)(s_atoms + f);
    __builtin_amdgcn_global_load_async_to_lds_b128(gsrc, ldst, 0, 0);
  }
  #if __has_builtin(__builtin_amdgcn_s_wait_asynccnt)
  __builtin_amdgcn_s_wait_asynccnt(0);
  #else
  asm volatile("s_wait_asynccnt 0" ::: "memory");
  #endif
#else
  // Fallback: plain vectorized staging
  for (int i = tid; i < (NB_A * 3) / 4; i += TPB)
    ((f32x4*)s_atoms)[i] = ((const f32x4*)atoms_b)[i];
#endif
  __syncthreads();

  const size_t BAN        = (size_t)NB_B * NB_A * NB_N;
  const size_t base_entry = ((size_t)b * NB_A + (size_t)chunk * APB) * NB_N;

  const int* nb_blk = neighbors + base_entry;
  const int* mk_blk = nmask     + base_entry;

  float* out_d = out;                                   // distances
  float* out_v = out + BAN;                             // vectors
  int*   out_n = (int*)out + (size_t)4 * BAN;           // neighbors passthrough
  int*   out_m = (int*)out + (size_t)5 * BAN;           // mask passthrough

  // APB*N = 8192 entries per block; 4 consecutive entries per thread per iter.
  #pragma unroll 2
  for (int it = 0; it < (APB * NB_N) / (TPB * 4); ++it) {   // 8 iterations
    const int e = (it * TPB + tid) * 4;                     // entry in block
    const int a = chunk * APB + (e >> 9);                   // e/512: same row for all 4
    const float cx = s_atoms[3 * a + 0];
    const float cy = s_atoms[3 * a + 1];
    const float cz = s_atoms[3 * a + 2];

    const i32x4 nb = *(const i32x4*)(nb_blk + e);           // coalesced b128
    const i32x4 mk = *(const i32x4*)(mk_blk + e);

    f32x4 dist;
    float vec[12];
    #pragma unroll
    for (int k = 0; k < 4; ++k) {
      const int j = nb[k];
      // LDS gather: stride-3 f32, coprime with 64 banks
      const float dx = s_atoms[3 * j + 0] - cx;
      const float dy = s_atoms[3 * j + 1] - cy;
      const float dz = s_atoms[3 * j + 2] - cz;
      const float d  = fast_sqrt(dx * dx + dy * dy + dz * dz);  // v_sqrt_f32
      const bool  m  = (mk[k] != 0);
      dist[k] = m ? d : 0.0f;
      // reference: tmp = where(mask, d+eps, 1.0); vec = dvec / tmp (unmasked!)
      const float inv = fast_rcp(m ? (d + EPS_F) : 1.0f);       // v_rcp_f32
      vec[3 * k + 0] = dx * inv;
      vec[3 * k + 1] = dy * inv;
      vec[3 * k + 2] = dz * inv;
    }

    const size_t g = base_entry + (size_t)e;
    __builtin_nontemporal_store(dist, (f32x4*)(out_d + g));

    float* vp = out_v + (size_t)3 * g;                      // 48B-aligned (g % 4 == 0)
    const f32x4 v0 = {vec[0], vec[1], vec[2],  vec[3]};
    const f32x4 v1 = {vec[4], vec[5], vec[6],  vec[7]};
    const f32x4 v2 = {vec[8], vec[9], vec[10], vec[11]};
    __builtin_nontemporal_store(v0, (f32x4*)(vp + 0));
    __builtin_nontemporal_store(v1, (f32x4*)(vp + 4));
    __builtin_nontemporal_store(v2, (f32x4*)(vp + 8));

    __builtin_nontemporal_store(nb, (i32x4*)(out_n + g));
    __builtin_nontemporal_store(mk, (i32x4*)(out_m + g));
  }
}

extern "C" void kernel_launch(void* const* d_in, const int* in_sizes, int n_in,
                              void* d_out, int out_size, void* d_ws, size_t ws_size,
                              hipStream_t stream) {
  (void)in_sizes; (void)n_in; (void)out_size; (void)d_ws; (void)ws_size;
  const float* atoms     = (const float*)d_in[0];
  const int*   neighbors = (const int*)d_in[1];
  const int*   nmask     = (const int*)d_in[2];

  dim3 grid(NB_B * (NB_A / APB));   // 2048 blocks
  dim3 block(TPB);                  // 256 threads = 8 waves
  neighbor_dist_kernel<<<grid, block, 0, stream>>>(atoms, neighbors, nmask, (float*)d_out);
}